// ProposalLayer_28922309771854
// MI455X (gfx1250) — compile-verified
//
#include <hip/hip_runtime.h>
#include <math.h>

typedef unsigned int u32;
typedef unsigned int u32x4 __attribute__((ext_vector_type(4)));
typedef int          i32x4 __attribute__((ext_vector_type(4)));
typedef int          i32x8 __attribute__((ext_vector_type(8)));

#define BATCHES   4
#define NPROP     16384
#define NCH       76
#define PER       12
#define NHB       12
#define PRE0      6300
#define PRE1      2700
#define POST0     358
#define POST1     154
#define NMS_THR   0.8f
#define W0        197          /* ceil(6300/32) suppression words per row */
#define W1        85           /* ceil(2700/32) */
#define TWO_PI_F  6.28318530717958647692f
#define PI_F      3.14159265358979323846f

// ---------------------------------------------------------------------------
// Kernel 1: bbox decode. One block decodes 128 consecutive rows of rpn_reg.
// The 128x76 f32 tile (38,912 B) is staged into LDS with a single
// TENSOR_LOAD_TO_LDS (TDM) issued by wave 0, then waited with s_wait_tensorcnt.
// ---------------------------------------------------------------------------
__global__ __launch_bounds__(128) void decode_kernel(
    const float* __restrict__ reg, const float* __restrict__ xyz,
    float* __restrict__ props)
{
  __shared__ float tile[128 * NCH];
  const int row0 = blockIdx.x * 128;
  const int tid  = threadIdx.x;

#if __has_builtin(__builtin_amdgcn_tensor_load_to_lds) && __has_builtin(__builtin_amdgcn_s_wait_tensorcnt)
  if (tid < 32) {  // uniform wave-level branch: TDM issues once per wave (EXEC ignored)
    unsigned long long ga = (unsigned long long)(const char*)(reg + (size_t)row0 * NCH);
    // Generic LDS pointer: aperture in addr[63:32], LDS byte offset in addr[31:0].
    u32 lds_off = (u32)(unsigned long long)&tile[0];
    // ---- D# group 0 (CDNA5 ISA) ----
    u32x4 g0;
    g0[0] = 1u;                                              // count=1 valid descriptor
    g0[1] = lds_off;                                         // lds_addr (bytes)
    g0[2] = (u32)ga;                                         // global_addr[31:0]
    g0[3] = ((u32)(ga >> 32) & 0x01FFFFFFu) | 0x80000000u;   // addr[56:32] | type=2
    // ---- D# group 1 (§8.4): data_size=4B, tensor 76x128, tile 76x128 ----
    i32x8 g1;
    g1[0] = (2 << 16);        // data_size = 4 bytes
    g1[1] = (NCH << 16);      // tensor_dim0 = 76 (bits 79:48 low half)
    g1[2] = (128 << 16);      // tensor_dim1 = 128 (bits 111:80 low half)
    g1[3] = (NCH << 16);      // tensor_dim1 hi = 0 | tile_dim0 = 76
    g1[4] = 128;              // tile_dim1 = 128, tile_dim2 = 0
    g1[5] = NCH;              // tensor_dim0_stride = 76
    g1[6] = 0;  g1[7] = 0;
    i32x4 gz = {0, 0, 0, 0};  // groups 2/3 unused (2-D tensor)
#if __clang_major__ >= 23
    i32x8 gz8 = {0, 0, 0, 0, 0, 0, 0, 0};
    __builtin_amdgcn_tensor_load_to_lds(g0, g1, gz, gz, gz8, 0);
#else
    __builtin_amdgcn_tensor_load_to_lds(g0, g1, gz, gz, 0);
#endif
    __builtin_amdgcn_s_wait_tensorcnt(0);
  }
#else
  for (int c = tid; c < 128 * NCH; c += 128)
    tile[c] = reg[(size_t)row0 * NCH + c];
#endif
  __syncthreads();

  const float* r  = &tile[tid * NCH];
  const int   row = row0 + tid;

  int xb = 0; float xm = r[0];
#pragma unroll
  for (int k = 1; k < PER; ++k) { float v = r[k];        if (v > xm) { xm = v; xb = k; } }
  int zb = 0; float zm = r[PER];
#pragma unroll
  for (int k = 1; k < PER; ++k) { float v = r[PER + k];  if (v > zm) { zm = v; zb = k; } }

  float x0 = xyz[(size_t)row * 3 + 0];
  float y0 = xyz[(size_t)row * 3 + 1];
  float z0 = xyz[(size_t)row * 3 + 2];

  float px = (float)xb * 0.5f + 0.25f - 3.0f + r[2 * PER + xb] * 0.5f + x0;
  float pz = (float)zb * 0.5f + 0.25f - 3.0f + r[3 * PER + zb] * 0.5f + z0;
  float py = y0 + r[4 * PER];

  int rb = 0; float rm = r[4 * PER + 1];
#pragma unroll
  for (int k = 1; k < NHB; ++k) { float v = r[4 * PER + 1 + k]; if (v > rm) { rm = v; rb = k; } }
  const float apc = TWO_PI_F / (float)NHB;
  float ry = (float)rb * apc + r[4 * PER + 1 + NHB + rb] * (apc * 0.5f);
  ry = fmodf(ry, TWO_PI_F);
  if (ry < 0.f)  ry += TWO_PI_F;
  if (ry > PI_F) ry -= TWO_PI_F;

  const float A0 = 1.52563191462f, A1 = 1.62856739989f, A2 = 3.88311640418f;
  float hh = r[73] * A0 + A0;
  float ww = r[74] * A1 + A1;
  float ll = r[75] * A2 + A2;
  py += 0.5f * hh;   // props[...,1] += props[...,3] * 0.5

  float* o = props + (size_t)row * 7;
  o[0] = px; o[1] = py; o[2] = pz; o[3] = hh; o[4] = ww; o[5] = ll; o[6] = ry;
}

// ---------------------------------------------------------------------------
// Kernel 2: per-batch bitonic sort of all 16384 (score, idx) pairs in LDS
// (132 KB — needs CDNA5's big WGP LDS), followed by per-range compaction
// (block-wide scan) producing top-pre_n scores / 7-dof boxes / BEV boxes.
// ---------------------------------------------------------------------------
__global__ __launch_bounds__(1024) void sort_compact_kernel(
    const float* __restrict__ scores, const float* __restrict__ props,
    float* __restrict__ cs, float* __restrict__ cp, float* __restrict__ bev,
    int* __restrict__ counts)
{
  __shared__ float s_score[NPROP];
  __shared__ u32   s_idx[NPROP];
  __shared__ int   s_scan[1024];

  const int b = blockIdx.x;
  const int t = threadIdx.x;

  for (int i = t; i < NPROP; i += 1024) { s_score[i] = scores[(size_t)b * NPROP + i]; s_idx[i] = (u32)i; }
  __syncthreads();

  // Bitonic sort, descending by score.
  for (u32 k = 2; k <= NPROP; k <<= 1) {
    for (u32 j = k >> 1; j > 0; j >>= 1) {
      for (u32 i = t; i < NPROP; i += 1024) {
        u32 x = i ^ j;
        if (x > i) {
          float a = s_score[i], c = s_score[x];
          bool descSeg = ((i & k) == 0);
          bool doSwap  = descSeg ? (a < c) : (a > c);
          if (doSwap) {
            s_score[i] = c; s_score[x] = a;
            u32 ti = s_idx[i]; s_idx[i] = s_idx[x]; s_idx[x] = ti;
          }
        }
      }
      __syncthreads();
    }
  }

  // Distance (sorted order) — thread t owns contiguous chunk [16t, 16t+16).
  const size_t pbase = (size_t)b * NPROP;
  float dloc[16];
#pragma unroll
  for (int e = 0; e < 16; ++e) {
    int i = t * 16 + e;
    dloc[e] = props[(pbase + s_idx[i]) * 7 + 2];
  }

  for (int rr = 0; rr < 2; ++rr) {
    const float lo  = rr ? 40.f : 0.f;
    const float hi  = rr ? 80.f : 40.f;
    const int   preN = rr ? PRE1 : PRE0;

    bool m[16]; int cnt = 0;
#pragma unroll
    for (int e = 0; e < 16; ++e) { m[e] = (dloc[e] > lo) && (dloc[e] <= hi); cnt += m[e] ? 1 : 0; }

    s_scan[t] = cnt;
    __syncthreads();
    for (int offn = 1; offn < 1024; offn <<= 1) {           // Hillis–Steele inclusive scan
      int v = (t >= offn) ? s_scan[t - offn] : 0;
      __syncthreads();
      s_scan[t] += v;
      __syncthreads();
    }
    int rank  = s_scan[t] - cnt;                            // exclusive prefix
    int total = s_scan[1023];

    float* csr  = cs  + (size_t)(b * 2 + rr) * PRE0;
    float* cpr  = cp  + (size_t)(b * 2 + rr) * PRE0 * 7;
    float* bevr = bev + (size_t)(b * 2 + rr) * PRE0 * 4;
#pragma unroll
    for (int e = 0; e < 16; ++e) {
      if (m[e]) {
        if (rank < preN) {
          int i = t * 16 + e;
          u32 oi = s_idx[i];
          const float* pr = props + (pbase + oi) * 7;
          float p0 = pr[0], p1 = pr[1], p2 = pr[2], p3 = pr[3], p4 = pr[4], p5 = pr[5], p6 = pr[6];
          csr[rank] = s_score[i];
          float* cq = cpr + (size_t)rank * 7;
          cq[0] = p0; cq[1] = p1; cq[2] = p2; cq[3] = p3; cq[4] = p4; cq[5] = p5; cq[6] = p6;
          float* bq = bevr + (size_t)rank * 4;                // (cx-w/2, cy-h/2, cx+w/2, cy+h/2)
          bq[0] = p0 - p3 * 0.5f; bq[1] = p2 - p5 * 0.5f;
          bq[2] = p0 + p3 * 0.5f; bq[3] = p2 + p5 * 0.5f;
        }
        rank++;
      }
    }
    if (t == 0) counts[b * 2 + rr] = (total < preN) ? total : preN;
    __syncthreads();
  }
}

// ---------------------------------------------------------------------------
// Kernel 3: parallel suppression-bitmask build (phase A of bitmask NMS).
// Row i gets W words; bit j set iff j>i, j<n, IoU(i,j) > 0.8.
// ---------------------------------------------------------------------------
__global__ __launch_bounds__(256) void nms_mask_kernel(
    const float* __restrict__ bev, const int* __restrict__ counts,
    u32* __restrict__ sup0, u32* __restrict__ sup1)
{
  __shared__ float4 s_box[256];
  const int pair = blockIdx.y;
  const int b  = pair >> 1;
  const int rr = pair & 1;
  const int preN = rr ? PRE1 : PRE0;
  const int W    = rr ? W1 : W0;
  if (blockIdx.x * 256 >= preN) return;   // uniform: unused blocks for range 1

  const int n   = counts[pair];
  const int row = blockIdx.x * 256 + threadIdx.x;
  const float* bevr = bev + (size_t)pair * PRE0 * 4;
  u32* sup = rr ? (sup1 + (size_t)b * PRE1 * W1) : (sup0 + (size_t)b * PRE0 * W0);

  const bool act = row < n;
  float4 my = make_float4(0.f, 0.f, 0.f, 0.f);
  float  myArea = 0.f;
  if (act) {
    my = *(const float4*)(bevr + (size_t)row * 4);
    myArea = (my.z - my.x) * (my.w - my.y);
  }

  for (int c0 = 0; c0 < preN; c0 += 256) {
    int j = c0 + threadIdx.x;
    float4 bx = make_float4(3e38f, 3e38f, 3e38f, 3e38f);
    if (j < n) bx = *(const float4*)(bevr + (size_t)j * 4);
    s_box[threadIdx.x] = bx;
    __syncthreads();

    if (act) {
      u32 wd[8];
#pragma unroll
      for (int k = 0; k < 8; ++k) wd[k] = 0u;
      int lim = preN - c0; if (lim > 256) lim = 256;
      for (int jj = 0; jj < lim; ++jj) {
        int jg = c0 + jj;
        if (jg > row && jg < n) {
          float4 o = s_box[jj];
          float iw = fmaxf(fminf(my.z, o.z) - fmaxf(my.x, o.x), 0.f);
          float ih = fmaxf(fminf(my.w, o.w) - fmaxf(my.y, o.y), 0.f);
          float inter = iw * ih;
          float oa = (o.z - o.x) * (o.w - o.y);
          float iou = inter / (myArea + oa - inter + 1e-8f);
          if (iou > NMS_THR) wd[jj >> 5] |= 1u << (jj & 31);
        }
      }
      int wbase = row * W + (c0 >> 5);
      int nw = (lim + 31) >> 5;
      for (int k = 0; k < nw; ++k) sup[wbase + k] = wd[k];
    }
    __syncthreads();
  }
}

// ---------------------------------------------------------------------------
// Kernel 4: sequential scan (phase B). One wave per (batch, range): keeps a
// removed-bitmask in LDS, ORs in the suppression row of each survivor, writes
// the first post_n kept boxes/scores to the output (zero-filled first).
// ---------------------------------------------------------------------------
__global__ __launch_bounds__(32) void nms_scan_kernel(
    const float* __restrict__ cs, const float* __restrict__ cp,
    const u32* __restrict__ sup0, const u32* __restrict__ sup1,
    const int* __restrict__ counts, float* __restrict__ out)
{
  __shared__ u32 remv[W0];
  const int pair = blockIdx.x;
  const int b  = pair >> 1;
  const int rr = pair & 1;
  const int n     = counts[pair];
  const int postN = rr ? POST1 : POST0;
  const int W     = rr ? W1 : W0;
  const u32* sup  = rr ? (sup1 + (size_t)b * PRE1 * W1) : (sup0 + (size_t)b * PRE0 * W0);
  const float* csr = cs + (size_t)pair * PRE0;
  const float* cpr = cp + (size_t)pair * PRE0 * 7;
  const int outRow0 = b * 512 + (rr ? POST0 : 0);
  float* obox = out;                          // ret_bbox3d: [B*512*7]
  float* oscr = out + BATCHES * 512 * 7;      // ret_scores: [B*512]
  const int lane = threadIdx.x;

  for (int i = lane; i < W; i += 32) remv[i] = 0u;
  for (int i = lane; i < postN * 7; i += 32) obox[(size_t)outRow0 * 7 + i] = 0.f;
  for (int i = lane; i < postN;     i += 32) oscr[outRow0 + i] = 0.f;
  __syncthreads();

  int kept = 0;
  for (int i = 0; i < n && kept < postN; ++i) {
    u32 rb = remv[i >> 5];
    if (!((rb >> (i & 31)) & 1u)) {
      const u32* srow = sup + (size_t)i * W;
      for (int w = lane; w < W; w += 32) remv[w] |= srow[w];
      if (lane < 7)
        obox[(size_t)(outRow0 + kept) * 7 + lane] = cpr[(size_t)i * 7 + lane];
      if (lane == 7)
        oscr[outRow0 + kept] = csr[i];
      kept++;
    }
    __syncthreads();
  }
}

// ---------------------------------------------------------------------------
extern "C" void kernel_launch(void* const* d_in, const int* in_sizes, int n_in,
                              void* d_out, int out_size, void* d_ws, size_t ws_size,
                              hipStream_t stream)
{
  (void)in_sizes; (void)n_in; (void)out_size; (void)ws_size;
  const float* rpn_scores = (const float*)d_in[0];
  const float* rpn_reg    = (const float*)d_in[1];
  const float* xyz        = (const float*)d_in[2];
  float* out = (float*)d_out;

  char* ws = (char*)d_ws;
  size_t off = 0;
  auto bump = [&](size_t bytes) -> void* {
    void* p = ws + off;
    off += (bytes + 255) & ~(size_t)255;
    return p;
  };
  float* props  = (float*)bump((size_t)BATCHES * NPROP * 7 * sizeof(float));   // 1.75 MB
  float* cs     = (float*)bump((size_t)BATCHES * 2 * PRE0 * sizeof(float));    // 0.20 MB
  float* cp     = (float*)bump((size_t)BATCHES * 2 * PRE0 * 7 * sizeof(float));// 1.41 MB
  float* bev    = (float*)bump((size_t)BATCHES * 2 * PRE0 * 4 * sizeof(float));// 0.81 MB
  int*   counts = (int*)  bump(8 * sizeof(int));
  u32*   sup0   = (u32*)  bump((size_t)BATCHES * PRE0 * W0 * sizeof(u32));     // 19.9 MB
  u32*   sup1   = (u32*)  bump((size_t)BATCHES * PRE1 * W1 * sizeof(u32));     // 3.7 MB

  decode_kernel<<<dim3((BATCHES * NPROP) / 128), dim3(128), 0, stream>>>(rpn_reg, xyz, props);
  sort_compact_kernel<<<dim3(BATCHES), dim3(1024), 0, stream>>>(rpn_scores, props, cs, cp, bev, counts);
  nms_mask_kernel<<<dim3((PRE0 + 255) / 256, BATCHES * 2), dim3(256), 0, stream>>>(bev, counts, sup0, sup1);
  nms_scan_kernel<<<dim3(BATCHES * 2), dim3(32), 0, stream>>>(cs, cp, sup0, sup1, counts, out);
}